// MiniBatchDiscrimination_51926154609300
// MI455X (gfx1250) — compile-verified
//
#include <hip/hip_runtime.h>

typedef __attribute__((ext_vector_type(2))) float v2f;
typedef __attribute__((ext_vector_type(8))) float v8f;

#define B_SZ   512
#define IN_F   1024
#define OUT_F  100
#define KD     5
#define OK     500      // OUT_F * KD
#define PLD    512      // padded leading dim of proj scratch

// ---------------------------------------------------------------------------
// Kernel 1: proj[B, OK(pad 512)] = x[B, IN] @ W[OK, IN]^T
// One wave per 16x16 output tile, V_WMMA_F32_16X16X4_F32, K swept in steps of 4.
// ---------------------------------------------------------------------------
__global__ __launch_bounds__(256) void mbd_gemm(const float* __restrict__ x,
                                                const float* __restrict__ W,
                                                float* __restrict__ proj) {
    const int lane = threadIdx.x & 31;
    const int wave = threadIdx.x >> 5;
    const int tile = blockIdx.x * 8 + wave;   // 0..1023
    const int tm   = tile >> 5;               // batch-row tile  (32 tiles)
    const int tn   = tile & 31;               // feature-col tile (32 tiles, padded)

    const int m    = lane & 15;
    const int koff = (lane >> 4) << 1;        // 0 or 2 (f32 A/B VGPR layout)

    const int arow = tm * 16 + m;             // row of x
    int       wrow = tn * 16 + m;             // row of W == output column
    if (wrow >= OK) wrow = OK - 1;            // clamp: padding cols never read back

    const float* ap = x + arow * IN_F + koff;
    const float* bp = W + wrow * IN_F + koff;

    v8f c0 = {}; v8f c1 = {};
    #pragma unroll 4
    for (int k = 0; k < IN_F; k += 8) {
        v2f a0 = *(const v2f*)(ap + k);
        v2f b0 = *(const v2f*)(bp + k);
        v2f a1 = *(const v2f*)(ap + k + 4);
        v2f b1 = *(const v2f*)(bp + k + 4);
        c0 = __builtin_amdgcn_wmma_f32_16x16x4_f32(false, a0, false, b0,
                                                   (short)0, c0, false, false);
        c1 = __builtin_amdgcn_wmma_f32_16x16x4_f32(false, a1, false, b1,
                                                   (short)0, c1, false, false);
    }
    v8f c = c0 + c1;   // elementwise: C/D layout is per-element, so this is exact

    // C/D layout: lane L, VGPR r -> (row = r + 8*(L>>4), col = L&15)
    const int  n     = lane & 15;
    const int  rbase = tm * 16 + 8 * (lane >> 4);
    float*     op    = proj + rbase * PLD + tn * 16 + n;
    #pragma unroll
    for (int r = 0; r < 8; ++r) op[r * PLD] = c[r];
}

// ---------------------------------------------------------------------------
// Kernel 2: mbd[i,o] = sum_j exp(-sum_k |p[i,o,k]-p[j,o,k]|) - 1
// block = (128,4): 4 batch rows, o across threadIdx.x (o<100 active).
// j staged through LDS in tiles of 32 rows (32*500*4 = 62.5 KB).
// ---------------------------------------------------------------------------
#define JT 32
__global__ __launch_bounds__(512) void mbd_pairs(const float* __restrict__ proj,
                                                 float* __restrict__ out) {
    __shared__ float sp[JT * OK];

    const int o   = threadIdx.x;                        // 0..127 (active < 100)
    const int i   = blockIdx.x * 4 + threadIdx.y;       // batch row
    const int tid = threadIdx.y * 128 + threadIdx.x;    // 0..511
    const bool act = (o < OUT_F);

    float pi[KD];
    if (act) {
        #pragma unroll
        for (int k = 0; k < KD; ++k) pi[k] = proj[i * PLD + o * KD + k];
    }

    float acc = 0.0f;
    for (int j0 = 0; j0 < B_SZ; j0 += JT) {
        __syncthreads();
        // cooperative stage: 32 rows x 500 cols, coalesced per row
        #pragma unroll 4
        for (int jj = 0; jj < JT; ++jj) {
            if (tid < OK) sp[jj * OK + tid] = proj[(j0 + jj) * PLD + tid];
        }
        __syncthreads();
        if (act) {
            #pragma unroll 4
            for (int jj = 0; jj < JT; ++jj) {
                const float* pj = sp + jj * OK + o * KD;
                float l1 = fabsf(pi[0] - pj[0]) + fabsf(pi[1] - pj[1])
                         + fabsf(pi[2] - pj[2]) + fabsf(pi[3] - pj[3])
                         + fabsf(pi[4] - pj[4]);
                acc += __expf(-l1);                     // v_exp_f32
            }
        }
    }
    if (act) out[i * (IN_F + OUT_F) + IN_F + o] = acc - 1.0f;
}

// ---------------------------------------------------------------------------
// Kernel 3: out[:, :1024] = x   (float4 copy; row stride 1124*4B is 16B-aligned)
// ---------------------------------------------------------------------------
__global__ __launch_bounds__(256) void mbd_copyx(const float* __restrict__ x,
                                                 float* __restrict__ out) {
    const int idx = blockIdx.x * 256 + threadIdx.x;     // float4 index into x
    const int row = idx / (IN_F / 4);
    const int col = idx % (IN_F / 4);
    float4 v = ((const float4*)x)[idx];
    ((float4*)(out + row * (IN_F + OUT_F)))[col] = v;
}

// ---------------------------------------------------------------------------
extern "C" void kernel_launch(void* const* d_in, const int* in_sizes, int n_in,
                              void* d_out, int out_size, void* d_ws, size_t ws_size,
                              hipStream_t stream) {
    const float* x = (const float*)d_in[0];   // [512, 1024]
    const float* W = (const float*)d_in[1];   // [500, 1024]
    float* out  = (float*)d_out;              // [512, 1124]
    float* proj = (float*)d_ws;               // [512, 512] fp32 scratch (1 MB)

    // 1024 wave-tiles, 8 waves (256 thr) per block -> 128 blocks
    mbd_gemm<<<128, 256, 0, stream>>>(x, W, proj);
    // 512 rows / 4 per block -> 128 blocks
    mbd_pairs<<<dim3(128), dim3(128, 4), 0, stream>>>(proj, out);
    // 512*1024/4 float4's / 256 -> 512 blocks
    mbd_copyx<<<512, 256, 0, stream>>>(x, out);
}